// Crude_Diag_16664473108760
// MI455X (gfx1250) — compile-verified
//
#include <hip/hip_runtime.h>

// y[t,j] = x[t,j] * diag(W)[j]  — exact f32, bandwidth-bound (~256 MiB @ 23.3 TB/s).
// Per-tile math runs on the CDNA5 matrix pipe: y_tile = x_tile @ D_block via
// v_wmma_f32_16x16x4_f32, with diagonal B chunks synthesized in registers.

#define N_FEAT 4096
#define N_TOK  8192

typedef __attribute__((ext_vector_type(2))) float v2f;  // 16x4 f32 A/B operand: 2 VGPRs
typedef __attribute__((ext_vector_type(8))) float v8f;  // 16x16 f32 C/D: 8 VGPRs

__global__ void Diag_Extract_kernel(const float* __restrict__ W, float* __restrict__ d) {
    int j = blockIdx.x * blockDim.x + threadIdx.x;
    if (j < N_FEAT) d[j] = W[(long)j * N_FEAT + j];
}

__global__ __launch_bounds__(256) void Diag_Scale_WMMA_kernel(
    const float* __restrict__ x, const float* __restrict__ d, float* __restrict__ y)
{
    const int lane = threadIdx.x & 31;   // wave32
    const int wave = threadIdx.x >> 5;   // 8 waves / block
    const int half = lane >> 4;          // half-wave select
    const int l16  = lane & 15;

    const int   fb = blockIdx.x << 4;    // feature block base: 0..4080
    const float dv = d[fb + l16];        // diag value for this lane's output column

    // B chunks: 4x16 slices of the 16x16 diagonal block D, built in registers.
    // Assumed f32 B layout (mirrors A 16x4): VGPR0 holds K = 2*half, VGPR1 holds K = 2*half+1,
    // N = l16.  B[k][n] = dv iff n == (4c + k), else 0.
    v2f b[4];
#pragma unroll
    for (int c = 0; c < 4; ++c) {
        const int kx = 2 * half;
        const int ky = 2 * half + 1;
        b[c].x = (l16 == 4 * c + kx) ? dv : 0.0f;
        b[c].y = (l16 == 4 * c + ky) ? dv : 0.0f;
    }

    // 128 waves per feature block (grid.y=16 * 8 waves), each owns 4 token tiles.
    const int wid = (blockIdx.y << 3) + wave;   // 0..127
#pragma unroll
    for (int i = 0; i < 4; ++i) {
        const int  tile = wid + (i << 7);       // 0..511 token tiles
        const long tb   = (long)tile << 4;      // token base row

        // A chunk c (16x4, f32): lanes 0-15 hold K=0,1 (VGPR0,1); lanes 16-31 hold K=2,3.
        // Lane reads a contiguous pair -> one aligned global_load_b64 per chunk.
        const float* xrow = x + (tb + l16) * (long)N_FEAT + fb + 2 * half;

        v8f acc = {};
#pragma unroll
        for (int c = 0; c < 4; ++c) {
            v2f a = *(const v2f*)(xrow + 4 * c);
            acc = __builtin_amdgcn_wmma_f32_16x16x4_f32(
                /*neg_a=*/false, a, /*neg_b=*/false, b[c],
                /*c_mod=*/(short)0, acc, /*reuse_a=*/false, /*reuse_b=*/false);
        }

        // C/D layout: VGPR v -> row M = v + 8*half, col N = l16.
        // Streaming output: non-temporal stores (no reuse, keep L2 for shared x lines).
#pragma unroll
        for (int v = 0; v < 8; ++v) {
            const long row = tb + v + 8 * half;
            __builtin_nontemporal_store(acc[v], y + row * (long)N_FEAT + fb + l16);
        }
    }
}

extern "C" void kernel_launch(void* const* d_in, const int* in_sizes, int n_in,
                              void* d_out, int out_size, void* d_ws, size_t ws_size,
                              hipStream_t stream) {
    const float* x = (const float*)d_in[0];          // [8192, 4096] f32
    const float* W = (const float*)d_in[1];          // [4096, 4096] f32 (diagonal)
    float*       y = (float*)d_out;                  // [8192, 4096] f32
    float*       diag = (float*)d_ws;                // 4096 f32 scratch (16 KiB)

    Diag_Extract_kernel<<<dim3(N_FEAT / 256), dim3(256), 0, stream>>>(W, diag);
    Diag_Scale_WMMA_kernel<<<dim3(N_FEAT / 16, 16), dim3(256), 0, stream>>>(x, diag, y);
}